// MultiSetAttention_36498632081555
// MI455X (gfx1250) — compile-verified
//
#include <hip/hip_runtime.h>
#include <math.h>

// Problem constants (B, N, F_IN=F_OUT=128 from the reference)
#define BATCH   8
#define NTOK    2048
#define FDIM    128
#define BN      (BATCH * NTOK)            // 16384 rows
#define BNF     ((size_t)BN * FDIM)       // 2,097,152 elements per (B,N,F) tensor
#define ALPHA_  0.1f
#define LDSTR   136                       // LDS row stride (floats): 2*136 % 64 == 16 -> halves hit disjoint banks
#define CH      32                        // attention key-chunk (keys per LDS stage)

typedef __attribute__((ext_vector_type(2))) float v2f;
typedef __attribute__((ext_vector_type(8))) float v8f;

__device__ __forceinline__ float lrelu(float x) { return x > 0.f ? x : ALPHA_ * x; }

// ---------------------------------------------------------------------------
// Kernel 1: Out(16384x128) = X(16384x128) @ W(128x128), fp32 WMMA 16x16x4.
// 4 waves / block, each wave owns a 16x128 output tile. W staged in LDS in
// 64-row chunks with padded stride to avoid bank conflicts on the split-half
// B-fragment reads (rows k+koff and k+koff+1).
// ---------------------------------------------------------------------------
__global__ __launch_bounds__(128)
void gemm_xw_wmma(const float* __restrict__ X, const float* __restrict__ W,
                  float* __restrict__ Out) {
  __shared__ float lw[64 * LDSTR];
  const int tid  = threadIdx.x;
  const int lane = tid & 31;
  const int wave = tid >> 5;
  const int l16  = lane & 15;
  const int half = lane >> 4;
  const int koff = half * 2;                 // K-pair owned by this lane-half
  const int m0   = blockIdx.x * 64 + wave * 16;

  v8f acc[8];
#pragma unroll
  for (int n = 0; n < 8; ++n) acc[n] = (v8f)(0.0f);

  for (int kc = 0; kc < FDIM; kc += 64) {
    __syncthreads();
    for (int i = tid; i < 64 * (FDIM / 4); i += 128) {   // 2048 float4
      const int row = i >> 5, c4 = i & 31;
      *(float4*)&lw[row * LDSTR + c4 * 4] =
          *(const float4*)&W[(size_t)(kc + row) * FDIM + c4 * 4];
    }
    __syncthreads();
    for (int k = 0; k < 64; k += 4) {
      const float2 av = *(const float2*)&X[(size_t)(m0 + l16) * FDIM + kc + k + koff];
      v2f a; a.x = av.x; a.y = av.y;
#pragma unroll
      for (int n = 0; n < 8; ++n) {
        v2f b;
        b.x = lw[(k + koff)     * LDSTR + n * 16 + l16];
        b.y = lw[(k + koff + 1) * LDSTR + n * 16 + l16];
        acc[n] = __builtin_amdgcn_wmma_f32_16x16x4_f32(
            false, a, false, b, (short)0, acc[n], false, false);
      }
    }
  }
#pragma unroll
  for (int n = 0; n < 8; ++n)
#pragma unroll
    for (int r = 0; r < 8; ++r)
      Out[(size_t)(m0 + r + half * 8) * FDIM + n * 16 + l16] = acc[n][r];
}

// ---------------------------------------------------------------------------
// Kernel 2: out[row] = dot(X[row,:128], w[:128]); one wave per row (wave32).
// ---------------------------------------------------------------------------
__global__ __launch_bounds__(256)
void rowdot(const float* __restrict__ X, const float* __restrict__ w,
            float* __restrict__ out) {
  const int lane = threadIdx.x & 31;
  const int wave = threadIdx.x >> 5;
  const int row  = blockIdx.x * 8 + wave;
  const float4 xv = ((const float4*)(X + (size_t)row * FDIM))[lane];
  const float4 wv = ((const float4*)w)[lane];
  float s = xv.x * wv.x + xv.y * wv.y + xv.z * wv.z + xv.w * wv.w;
#pragma unroll
  for (int off = 16; off; off >>= 1) s += __shfl_xor(s, off, 32);
  if (lane == 0) out[row] = s;
}

// ---------------------------------------------------------------------------
// Kernel 3: softmax stats. Scores are rank-1: e[q,k]=lrelu(s_q[q]+s_k[k]) with
// the friend-block diagonal excluded. One wave per query computes m, Z.
// ---------------------------------------------------------------------------
__global__ __launch_bounds__(256)
void softmax_stats(const float* __restrict__ sq1, const float* __restrict__ sq2,
                   const float* __restrict__ skP1, const float* __restrict__ skP2,
                   const float* __restrict__ skQ1, const float* __restrict__ skQ2,
                   float* __restrict__ m1, float* __restrict__ z1,
                   float* __restrict__ m2, float* __restrict__ z2) {
  const int lane = threadIdx.x & 31;
  const int wave = threadIdx.x >> 5;
  const int q    = blockIdx.x * 8 + wave;
  const int b    = blockIdx.y;
  const int set  = blockIdx.z;

  const float* skc = (set == 0 ? skP2 : skP1) + (size_t)b * NTOK;  // cross block keys
  const float* skf = (set == 0 ? skQ1 : skQ2) + (size_t)b * NTOK;  // friend block keys
  const float  sqv = (set == 0 ? sq1 : sq2)[(size_t)b * NTOK + q];

  float m = -3.0e38f, z = 0.f;
  for (int k = lane; k < 2 * NTOK; k += 32) {
    const bool isF = k >= NTOK;
    if (isF && (k - NTOK == q)) continue;                 // masked diagonal
    const float skv = isF ? skf[k - NTOK] : skc[k];
    const float s   = lrelu(sqv + skv);
    const float nm  = fmaxf(m, s);
    z = z * __expf(m - nm) + __expf(s - nm);
    m = nm;
  }
#pragma unroll
  for (int off = 16; off; off >>= 1) {
    const float om = __shfl_xor(m, off, 32);
    const float oz = __shfl_xor(z, off, 32);
    const float nm = fmaxf(m, om);
    z = z * __expf(m - nm) + oz * __expf(om - nm);
    m = nm;
  }
  if (lane == 0) {
    (set == 0 ? m1 : m2)[(size_t)b * NTOK + q] = m;
    (set == 0 ? z1 : z2)[(size_t)b * NTOK + q] = z;
  }
}

// ---------------------------------------------------------------------------
// Kernel 4: out = att @ h + x@Ws, fp32 WMMA, with double-buffered async
// global->LDS staging (GLOBAL_LOAD_ASYNC_TO_LDS_B128, ASYNCcnt-tracked) so the
// prefetch of key-chunk c+1 overlaps the 64 WMMAs of chunk c.
//
// `smem` is the ONLY shared object in this kernel, so its LDS offset is 0 and
// raw LDS byte addresses for the async copies are element offsets * 4.
// Layout: hbuf0 [0, CH*LDSTR), hbuf1 [CH*LDSTR, 2*CH*LDSTR),
//         skbuf0/skbuf1 at 2*CH*LDSTR (+CH).
// ---------------------------------------------------------------------------
#define SMEM_H1  (CH * LDSTR)
#define SMEM_SK  (2 * CH * LDSTR)
#define SMEM_TOT (2 * CH * LDSTR + 2 * CH)

__device__ __forceinline__ void stage_chunk_async(const float* __restrict__ hsrc,
                                                  const float* __restrict__ sks,
                                                  float* smem, int hbase, int skbase,
                                                  int tid) {
  if (tid < CH) smem[skbase + tid] = sks[tid];            // s_k: tiny, normal path
  // CH rows x 128 floats = CH*32 float4; 256 threads -> 4 async b128 per thread.
#pragma unroll
  for (int j = 0; j < (CH * (FDIM / 4)) / 256; ++j) {
    const int i   = tid + j * 256;
    const int row = i >> 5, c4 = i & 31;
    const float* gp    = hsrc + (size_t)row * FDIM + c4 * 4;
    const unsigned ldsb = (unsigned)((hbase + row * LDSTR + c4 * 4) * 4);
    asm volatile("global_load_async_to_lds_b128 %0, %1, off"
                 :: "v"(ldsb), "v"(gp) : "memory");
  }
}

__global__ __launch_bounds__(256)
void attn_out_wmma(const float* __restrict__ P1, const float* __restrict__ P2,
                   const float* __restrict__ Q1, const float* __restrict__ Q2,
                   const float* __restrict__ S1, const float* __restrict__ S2,
                   const float* __restrict__ sq1, const float* __restrict__ sq2,
                   const float* __restrict__ skP1, const float* __restrict__ skP2,
                   const float* __restrict__ skQ1, const float* __restrict__ skQ2,
                   const float* __restrict__ m1, const float* __restrict__ z1,
                   const float* __restrict__ m2, const float* __restrict__ z2,
                   float* __restrict__ out) {
  __shared__ __align__(16) float smem[SMEM_TOT];
  const int tid  = threadIdx.x;
  const int lane = tid & 31;
  const int wave = tid >> 5;
  const int l16  = lane & 15;
  const int half = lane >> 4;
  const int koff = half * 2;
  const int b    = blockIdx.y;
  const int set  = blockIdx.z;
  const int q0   = blockIdx.x * 128 + wave * 16;

  const float* hc  = (set == 0 ? P2 : P1) + (size_t)b * NTOK * FDIM; // cross rows of h
  const float* hf  = (set == 0 ? Q1 : Q2) + (size_t)b * NTOK * FDIM; // friend rows of h
  const float* skc = (set == 0 ? skP2 : skP1) + (size_t)b * NTOK;
  const float* skf = (set == 0 ? skQ1 : skQ2) + (size_t)b * NTOK;
  const float* S   = (set == 0 ? S1 : S2);
  float* outp      = out + (size_t)set * BNF;

  const int   qg  = q0 + l16;                        // A-fragment row M = lane%16
  const float sqv = (set == 0 ? sq1 : sq2)[(size_t)b * NTOK + qg];
  const float mv  = (set == 0 ? m1 : m2)[(size_t)b * NTOK + qg];
  const float rz  = 1.0f / (set == 0 ? z1 : z2)[(size_t)b * NTOK + qg];

  v8f acc[8];
#pragma unroll
  for (int n = 0; n < 8; ++n) acc[n] = (v8f)(0.0f);

  // Prologue: start chunk 0 into buffer 0.
  {
    const float* hsrc = hc;                          // kc = 0 is always in the cross block
    const float* sks  = skc;
    stage_chunk_async(hsrc, sks, smem, 0, SMEM_SK, tid);
  }

  for (int kc = 0, c = 0; kc < 2 * NTOK; kc += CH, ++c) {
    // Prefetch chunk c+1 into the other buffer, then wait for chunk c's data.
    const int kcn = kc + CH;
    if (kcn < 2 * NTOK) {
      const bool nF = kcn >= NTOK;
      const float* hsrc = nF ? (hf + (size_t)(kcn - NTOK) * FDIM) : (hc + (size_t)kcn * FDIM);
      const float* sks  = nF ? (skf + (kcn - NTOK)) : (skc + kcn);
      const int nb = (c + 1) & 1;
      stage_chunk_async(hsrc, sks, smem, nb * SMEM_H1, SMEM_SK + nb * CH, tid);
      // 4 async instrs just issued for c+1; in-order completion => chunk c done.
      asm volatile("s_wait_asynccnt 4" ::: "memory");
    } else {
      asm volatile("s_wait_asynccnt 0" ::: "memory");
    }
    __syncthreads();                                 // publish chunk c across waves

    const int  hb   = (c & 1) * SMEM_H1;
    const int  skb  = SMEM_SK + (c & 1) * CH;
    const bool isF  = kc >= NTOK;
    const int  diag = isF ? (qg - (kc - NTOK)) : -1; // chunk-local masked key for this lane's row

    for (int k = 0; k < CH; k += 4) {
      const float sk0 = smem[skb + k + koff];
      const float sk1 = smem[skb + k + koff + 1];
      const float w0  = __expf(lrelu(sqv + sk0) - mv) * rz;
      const float w1  = __expf(lrelu(sqv + sk1) - mv) * rz;
      v2f a;
      a.x = (k + koff     == diag) ? 0.f : w0;       // select, not branch: EXEC stays full
      a.y = (k + koff + 1 == diag) ? 0.f : w1;
#pragma unroll
      for (int n = 0; n < 8; ++n) {
        v2f bf;
        bf.x = smem[hb + (k + koff)     * LDSTR + n * 16 + l16];
        bf.y = smem[hb + (k + koff + 1) * LDSTR + n * 16 + l16];
        acc[n] = __builtin_amdgcn_wmma_f32_16x16x4_f32(
            false, a, false, bf, (short)0, acc[n], false, false);
      }
    }
    __syncthreads();   // all waves done reading buf[c&1] before it is overwritten
  }

#pragma unroll
  for (int n = 0; n < 8; ++n)
#pragma unroll
    for (int r = 0; r < 8; ++r) {
      const size_t idx = ((size_t)b * NTOK + q0 + r + half * 8) * FDIM + n * 16 + l16;
      outp[idx] = acc[n][r] + S[idx];
    }
}

// ---------------------------------------------------------------------------
extern "C" void kernel_launch(void* const* d_in, const int* in_sizes, int n_in,
                              void* d_out, int out_size, void* d_ws, size_t ws_size,
                              hipStream_t stream) {
  const float* x1  = (const float*)d_in[0];
  const float* x2  = (const float*)d_in[1];
  const float* Wc  = (const float*)d_in[2];
  const float* Wf  = (const float*)d_in[3];
  const float* Ws  = (const float*)d_in[4];
  const float* att = (const float*)d_in[5];   // (256,1): a_q = [0:128), a_k = [128:256)
  const float* a_q = att;
  const float* a_k = att + FDIM;

  float* ws   = (float*)d_ws;                 // needs 6*BNF + 10*BN floats (~51 MB)
  float* P1   = ws + 0 * BNF;                 // x1 @ Wc
  float* P2   = ws + 1 * BNF;                 // x2 @ Wc
  float* Q1   = ws + 2 * BNF;                 // x1 @ Wf
  float* Q2   = ws + 3 * BNF;                 // x2 @ Wf
  float* S1   = ws + 4 * BNF;                 // x1 @ Ws
  float* S2   = ws + 5 * BNF;                 // x2 @ Ws
  float* sq1  = ws + 6 * BNF;
  float* sq2  = sq1 + 1 * BN;
  float* skP1 = sq1 + 2 * BN;
  float* skP2 = sq1 + 3 * BN;
  float* skQ1 = sq1 + 4 * BN;
  float* skQ2 = sq1 + 5 * BN;
  float* m1   = sq1 + 6 * BN;
  float* z1   = sq1 + 7 * BN;
  float* m2   = sq1 + 8 * BN;
  float* z2   = sq1 + 9 * BN;
  float* outp = (float*)d_out;                // out1 then out2, each BNF floats

  // Phase 1: six 16384x128x128 GEMMs on the fp32 WMMA pipe
  gemm_xw_wmma<<<BN / 64, 128, 0, stream>>>(x1, Wc, P1);
  gemm_xw_wmma<<<BN / 64, 128, 0, stream>>>(x2, Wc, P2);
  gemm_xw_wmma<<<BN / 64, 128, 0, stream>>>(x1, Wf, Q1);
  gemm_xw_wmma<<<BN / 64, 128, 0, stream>>>(x2, Wf, Q2);
  gemm_xw_wmma<<<BN / 64, 128, 0, stream>>>(x1, Ws, S1);
  gemm_xw_wmma<<<BN / 64, 128, 0, stream>>>(x2, Ws, S2);

  // Phase 2: attention score vectors (rank-1 structure)
  rowdot<<<BN / 8, 256, 0, stream>>>(Q1, a_q, sq1);
  rowdot<<<BN / 8, 256, 0, stream>>>(Q2, a_q, sq2);
  rowdot<<<BN / 8, 256, 0, stream>>>(P1, a_k, skP1);
  rowdot<<<BN / 8, 256, 0, stream>>>(P2, a_k, skP2);
  rowdot<<<BN / 8, 256, 0, stream>>>(Q1, a_k, skQ1);
  rowdot<<<BN / 8, 256, 0, stream>>>(Q2, a_k, skQ2);

  // Phase 3: per-query softmax max/partition (makes phase 4 rescale-free)
  softmax_stats<<<dim3(NTOK / 8, BATCH, 2), 256, 0, stream>>>(
      sq1, sq2, skP1, skP2, skQ1, skQ2, m1, z1, m2, z2);

  // Phase 4: out = att @ h + self, fp32 WMMA with pre-normalized weights and
  // double-buffered async global->LDS key/value streaming.
  attn_out_wmma<<<dim3(NTOK / 128, BATCH, 2), 256, 0, stream>>>(
      P1, P2, Q1, Q2, S1, S2, sq1, sq2, skP1, skP2, skQ1, skQ2,
      m1, z1, m2, z2, outp);
}